// LocalGlobalAttention_72851235275372
// MI455X (gfx1250) — compile-verified
//
#include <hip/hip_runtime.h>
#include <math.h>

typedef __attribute__((ext_vector_type(2))) float v2f;
typedef __attribute__((ext_vector_type(8))) float v8f;

#define NB 8
#define NC 64
#define NH 256
#define NW 256
#define PS 8
#define NHP 32            // patches per column (H/P)
#define NWP 32            // patches per row (W/P)
#define DIM 64            // P*P
#define HID 128
#define NP 16             // patches per workgroup
#define MEANSTR 68        // padded stride for s_mean (multiple of 4 -> 16B aligned rows)
#define HSTR 132          // padded stride for s_h
#define PLANE (NH*NW)     // 65536

__global__ void __launch_bounds__(256)
lga_phase1(const float* __restrict__ x,
           const float* __restrict__ w1,  const float* __restrict__ b1,
           const float* __restrict__ lng, const float* __restrict__ lnb,
           const float* __restrict__ w2,  const float* __restrict__ b2,
           const float* __restrict__ task,
           const float* __restrict__ cw,  const float* __restrict__ cb,
           float* __restrict__ out32)
{
    __shared__ float s_mean[NP * MEANSTR];   // [patch][pixel]
    __shared__ float s_h[NP * HSTR];         // [patch][hid]
    __shared__ float s_attn[NP * DIM];       // [patch][pixel]
    __shared__ float s_sel[DIM * NP];        // [channel][patch]  (tokens then selected)
    __shared__ float s_sim[NP];

    const int tid  = threadIdx.x;
    const int lane = tid & 31;
    const int wave = tid >> 5;
    const int lm   = lane & 15;
    const int koff = (lane < 16) ? 0 : 2;
    const int mbase = (lane < 16) ? 0 : 8;

    const int b   = blockIdx.x >> 6;
    const int rem = blockIdx.x & 63;
    const int li  = rem >> 1;
    const int lj0 = (rem & 1) * NP;

    // ------- mean over channels: one b128 per channel per thread (HBM read of x) ----
    {
        const size_t base = (((size_t)b * NC) * NH + (size_t)li * PS) * NW + (size_t)lj0 * PS;
        const int q = tid * 4;                       // 4 consecutive pixels, same patch
        const int off = (q >> 7) * NW + (q & 127);   // wave reads 512B contiguous
        float ax = 0.f, ay = 0.f, az = 0.f, aw = 0.f;
        const float* p = x + base + off;
        #pragma unroll 4
        for (int c = 0; c < NC; ++c) {
            const float4 v = *(const float4*)p;
            ax += v.x; ay += v.y; az += v.z; aw += v.w;
            p += PLANE;
        }
        const float inv = 1.0f / (float)NC;
        const int l   = (q & 127) >> 3;
        const int pix = (q >> 7) * 8 + (q & 7);      // multiple of 4 -> aligned
        float4 r; r.x = ax * inv; r.y = ay * inv; r.z = az * inv; r.w = aw * inv;
        *(float4*)&s_mean[l * MEANSTR + pix] = r;
    }
    __syncthreads();

    // ------- layer 1: [16x64] @ [64x128] via fp32 WMMA, wave j = N-tile j -----------
    {
        const int n = 16 * wave + lm;        // hidden index 0..127
        v8f acc = {};
        #pragma unroll
        for (int kk = 0; kk < DIM; kk += 4) {
            v2f A, B;
            A.x = s_mean[lm * MEANSTR + kk + koff];
            A.y = s_mean[lm * MEANSTR + kk + koff + 1];
            B.x = w1[n * DIM + kk + koff];
            B.y = w1[n * DIM + kk + koff + 1];
            acc = __builtin_amdgcn_wmma_f32_16x16x4_f32(false, A, false, B,
                                                        (short)0, acc, false, false);
        }
        const float bias = b1[n];
        #pragma unroll
        for (int r = 0; r < 8; ++r)
            s_h[(mbase + r) * HSTR + n] = acc[r] + bias;
    }
    __syncthreads();

    // ------- LayerNorm + exact GELU, one thread per patch row -----------------------
    if (tid < NP) {
        float* hr = &s_h[tid * HSTR];
        float mu = 0.f;
        for (int k = 0; k < HID; ++k) mu += hr[k];
        mu *= (1.0f / HID);
        float var = 0.f;
        for (int k = 0; k < HID; ++k) { float d = hr[k] - mu; var += d * d; }
        var *= (1.0f / HID);
        const float inv = rsqrtf(var + 1e-5f);
        for (int k = 0; k < HID; ++k) {
            float v = (hr[k] - mu) * inv * lng[k] + lnb[k];
            hr[k] = 0.5f * v * (1.0f + erff(v * 0.70710678118654752f));
        }
    }
    __syncthreads();

    // ------- layer 2: [16x128] @ [128x64] via fp32 WMMA, waves 0..3 -----------------
    if (wave < 4) {
        const int n = 16 * wave + lm;        // pixel logit index 0..63
        v8f acc = {};
        #pragma unroll
        for (int kk = 0; kk < HID; kk += 4) {
            v2f A, B;
            A.x = s_h[lm * HSTR + kk + koff];
            A.y = s_h[lm * HSTR + kk + koff + 1];
            B.x = w2[n * HID + kk + koff];
            B.y = w2[n * HID + kk + koff + 1];
            acc = __builtin_amdgcn_wmma_f32_16x16x4_f32(false, A, false, B,
                                                        (short)0, acc, false, false);
        }
        const float bias = b2[n];
        #pragma unroll
        for (int r = 0; r < 8; ++r)
            s_attn[(mbase + r) * DIM + n] = acc[r] + bias;
    }
    __syncthreads();

    // ------- softmax over 64 pixels, one thread per patch row -----------------------
    if (tid < NP) {
        float* ar = &s_attn[tid * DIM];
        float mx = ar[0];
        for (int k = 1; k < DIM; ++k) mx = fmaxf(mx, ar[k]);
        float s = 0.f;
        for (int k = 0; k < DIM; ++k) { float e = expf(ar[k] - mx); ar[k] = e; s += e; }
        const float inv = 1.0f / s;
        for (int k = 0; k < DIM; ++k) ar[k] *= inv;
    }
    __syncthreads();

    // ------- attention-weighted pooling (x re-read hits 192MB L2) -------------------
    #pragma unroll
    for (int i = 0; i < 4; ++i) {
        const int idx = tid + (i << 8);
        const int l = idx >> 6, c = idx & 63;
        const float* pp = x + ((((size_t)b * NC + c) * NH + (size_t)li * PS)) * NW
                            + (size_t)(lj0 + l) * PS;
        const float* aw = &s_attn[l * DIM];
        float t = 0.f;
        #pragma unroll
        for (int ki = 0; ki < 8; ++ki) {
            const float4 r0 = *(const float4*)(pp + ki * NW);
            const float4 r1 = *(const float4*)(pp + ki * NW + 4);
            const float* a = aw + ki * 8;
            t += r0.x * a[0] + r0.y * a[1] + r0.z * a[2] + r0.w * a[3]
               + r1.x * a[4] + r1.y * a[5] + r1.z * a[6] + r1.w * a[7];
        }
        s_sel[c * NP + l] = t;               // tokens[c][l]
    }
    __syncthreads();

    // ------- cosine similarity vs task embedding ------------------------------------
    if (tid < NP) {
        const int l = tid;
        float n2 = 0.f, dot = 0.f, t2 = 0.f;
        for (int c = 0; c < NC; ++c) {
            float tk = s_sel[c * NP + l];
            float te = task[c];
            n2 += tk * tk; dot += te * tk; t2 += te * te;
        }
        s_sim[l] = dot / (fmaxf(sqrtf(n2), 1e-12f) * fmaxf(sqrtf(t2), 1e-12f));
    }
    __syncthreads();

    // selected = tokens * sim
    #pragma unroll
    for (int i = 0; i < 4; ++i) {
        const int idx = tid + (i << 8);      // idx = c*16 + l
        s_sel[idx] *= s_sim[idx & 15];
    }
    __syncthreads();

    // ------- 1x1 conv: [64x64] @ [64x16] via fp32 WMMA, waves 0..3 ------------------
    if (wave < 4) {
        const int o = 16 * wave + lm;        // A-matrix row = output channel
        v8f acc = {};
        #pragma unroll
        for (int kk = 0; kk < NC; kk += 4) {
            v2f A, B;
            A.x = cw[o * NC + kk + koff];
            A.y = cw[o * NC + kk + koff + 1];
            B.x = s_sel[(kk + koff) * NP + lm];
            B.y = s_sel[(kk + koff + 1) * NP + lm];
            acc = __builtin_amdgcn_wmma_f32_16x16x4_f32(false, A, false, B,
                                                        (short)0, acc, false, false);
        }
        #pragma unroll
        for (int r = 0; r < 8; ++r) {
            const int orow = 16 * wave + mbase + r;
            out32[(((size_t)b * NC + orow) * NHP + li) * NWP + lj0 + lm] = acc[r] + cb[orow];
        }
    }
}

// -------- bilinear upsample 32x32 -> 256x256, half-pixel centers, one plane/block --
// Each thread owns a 4-wide x-block (b128 stores, 512B contiguous per wave) and
// strides over rows.
__global__ void __launch_bounds__(256)
lga_upsample(const float* __restrict__ src, float* __restrict__ dst)
{
    __shared__ float s[NHP * NWP];
    const int plane = blockIdx.x;
    const int tid = threadIdx.x;
    ((float4*)s)[tid] = ((const float4*)(src + (size_t)plane * (NHP * NWP)))[tid];
    __syncthreads();

    const int xb = (tid & 63) * 4;           // x block start (0..252)
    const int yb = tid >> 6;                 // row phase (0..3)

    int x0c[4], x1c[4];
    float wx[4];
    #pragma unroll
    for (int j = 0; j < 4; ++j) {
        const float sx = (xb + j) * 0.125f - 0.4375f;
        const int xf = (int)floorf(sx);
        wx[j] = sx - (float)xf;
        x0c[j] = (xf < 0) ? 0 : ((xf > 31) ? 31 : xf);
        const int x1i = xf + 1;
        x1c[j] = (x1i < 0) ? 0 : ((x1i > 31) ? 31 : x1i);
    }

    float* dp = dst + (size_t)plane * PLANE;
    for (int yy = 0; yy < 64; ++yy) {
        const int y = yy * 4 + yb;
        const float sy = y * 0.125f - 0.4375f;
        const int yf = (int)floorf(sy);
        const float wy = sy - (float)yf;
        const int y0 = (yf < 0) ? 0 : ((yf > 31) ? 31 : yf);
        const int y1i = yf + 1;
        const int y1 = (y1i < 0) ? 0 : ((y1i > 31) ? 31 : y1i);
        const float* r0 = s + y0 * NWP;
        const float* r1 = s + y1 * NWP;
        float4 o;
        float* op = (float*)&o;
        #pragma unroll
        for (int j = 0; j < 4; ++j) {
            const float a = r0[x0c[j]] + wx[j] * (r0[x1c[j]] - r0[x0c[j]]);
            const float c = r1[x0c[j]] + wx[j] * (r1[x1c[j]] - r1[x0c[j]]);
            op[j] = a + wy * (c - a);
        }
        *(float4*)(dp + y * NW + xb) = o;
    }
}

extern "C" void kernel_launch(void* const* d_in, const int* in_sizes, int n_in,
                              void* d_out, int out_size, void* d_ws, size_t ws_size,
                              hipStream_t stream) {
    const float* x    = (const float*)d_in[0];
    const float* w1   = (const float*)d_in[1];
    const float* b1   = (const float*)d_in[2];
    const float* lng  = (const float*)d_in[3];
    const float* lnb  = (const float*)d_in[4];
    const float* w2   = (const float*)d_in[5];
    const float* b2   = (const float*)d_in[6];
    const float* task = (const float*)d_in[7];
    const float* cw   = (const float*)d_in[8];
    const float* cb   = (const float*)d_in[9];
    float* out32 = (float*)d_ws;                       // 8*64*32*32 floats = 2 MB
    float* out   = (float*)d_out;

    lga_phase1<<<NB * NHP * (NWP / NP), 256, 0, stream>>>(
        x, w1, b1, lng, lnb, w2, b2, task, cw, cb, out32);
    lga_upsample<<<NB * NC, 256, 0, stream>>>(out32, out);
}